// RNNAgent2_26912265076966
// MI455X (gfx1250) — compile-verified
//
#include <hip/hip_runtime.h>
#include <hip/hip_bf16.h>
#include <stdint.h>

typedef __attribute__((ext_vector_type(2))) float v2f;
typedef __attribute__((ext_vector_type(8))) float v8f;

#define Bsz 64
#define Tn  1024
#define Kn  16
#define In  3
#define Hn  20
#define Gn  80      // 4*H gate columns
#define GS  84      // gates LDS row stride (floats), padded vs 80 to dodge bank conflicts
#define HS  20      // h LDS row stride (floats), even -> 8B-aligned v2f loads

__device__ __forceinline__ float ftanh(float x) {
#if __has_builtin(__builtin_amdgcn_tanhf)
    return __builtin_amdgcn_tanhf(x);                       // v_tanh_f32 (TRANS)
#else
    return 1.f - 2.f * __builtin_amdgcn_rcpf(__expf(2.f * x) + 1.f);
#endif
}
__device__ __forceinline__ float fsig(float x) {
#if __has_builtin(__builtin_amdgcn_tanhf)
    return fmaf(0.5f, __builtin_amdgcn_tanhf(0.5f * x), 0.5f);
#else
    return __builtin_amdgcn_rcpf(1.f + __expf(-x));         // v_exp + v_rcp
#endif
}

__global__ __launch_bounds__(160) void lstm16_wmma_kernel(
    const float* __restrict__ inp,   const float* __restrict__ wvec,
    const float* __restrict__ W_ih,  const float* __restrict__ W_hh,
    const float* __restrict__ b_ih,  const float* __restrict__ b_hh,
    const float* __restrict__ conv_w, const float* __restrict__ conv_b,
    float* __restrict__ logits)
{
    __shared__ float h_s[16 * HS];        // hidden state, rows = batch rows of this tile
    __shared__ float g_s[16 * GS];        // gate accumulators (i|f|g|o) per row
    __shared__ float x_s[4][16][4];       // 4-slot async ring for x_t (3 vals + zero pad)

    const int tid  = threadIdx.x;
    const int lane = tid & 31;
    const int wid  = tid >> 5;            // wave id 0..4 -> gate-column tile
    const int k    = blockIdx.x >> 2;     // LSTM index
    const int mt   = blockIdx.x & 3;      // 16-row batch tile

    const int n    = lane & 15;           // A-row (M) and B/C-column (N) for this lane
    const int lh   = lane >> 4;           // lane half
    const int kk0  = lh * 2;              // K offset within 4-chunk per ISA 7.12.2

    // ---- zero h0 and the x padding element -------------------------------
    for (int i = tid; i < 16 * HS; i += 160) h_s[i] = 0.f;
    if (tid < 64) x_s[tid >> 4][tid & 15][3] = 0.f;

    // ---- loop-invariant B fragments (weights) + bias in VGPRs ------------
    const float* Wh = W_hh + ((size_t)k * Gn + (wid * 16 + n)) * Hn;
    v2f bfr[5];
#pragma unroll
    for (int c = 0; c < 5; ++c) {
        bfr[c].x = Wh[4 * c + kk0];
        bfr[c].y = Wh[4 * c + kk0 + 1];
    }
    const float* Wx = W_ih + ((size_t)k * Gn + (wid * 16 + n)) * In;
    v2f bx;
    bx.x = (kk0 == 0) ? Wx[0] : Wx[2];
    bx.y = (kk0 == 0) ? Wx[1] : 0.f;
    const float biasv = b_ih[k * Gn + wid * 16 + n] + b_hh[k * Gn + wid * 16 + n];

    // ---- async-x prefetch machinery (wave 0 only) ------------------------
    const int r0 = lane & 15;             // lanes 16..31 mirror lanes 0..15 (benign dup)
    const uint32_t ldsx0 = (uint32_t)(uintptr_t)&x_s[0][r0][0];
    uint32_t gnext = (uint32_t)((((mt * 16 + r0) * Tn) * Kn + k) * In) * 4u;
    if (wid == 0) {
#pragma unroll
        for (int t0 = 0; t0 < 3; ++t0) {
            uint32_t la = ldsx0 + (uint32_t)t0 * 256u;
            asm volatile(
                "global_load_async_to_lds_b32 %0, %1, %2 offset:0\n\t"
                "global_load_async_to_lds_b32 %0, %1, %2 offset:4\n\t"
                "global_load_async_to_lds_b32 %0, %1, %2 offset:8"
                :: "v"(la), "v"(gnext), "s"(inp) : "memory");
            gnext += (uint32_t)(Kn * In * 4);
        }
    }

    // ---- per-thread c-state (2 LSTM cells / thread: 320 cells, 160 threads)
    float cst0 = 0.f, cst1 = 0.f;
    const int cell0 = tid * 2, cell1 = tid * 2 + 1;
    const int rr0 = cell0 / Hn, hh0 = cell0 % Hn;
    const int rr1 = cell1 / Hn, hh1 = cell1 % Hn;

    // One LSTM timestep. `prefetch` is compile-time; wait code passed per call.
    auto step = [&](int t, bool prefetch) {
        __syncthreads();   // x[t] visible to all waves; h from t-1 visible

        if (prefetch && wid == 0) {   // slot (t+3)&3 free after the barrier above
            uint32_t la = ldsx0 + (uint32_t)((t + 3) & 3) * 256u;
            asm volatile(
                "global_load_async_to_lds_b32 %0, %1, %2 offset:0\n\t"
                "global_load_async_to_lds_b32 %0, %1, %2 offset:4\n\t"
                "global_load_async_to_lds_b32 %0, %1, %2 offset:8"
                :: "v"(la), "v"(gnext), "s"(inp) : "memory");
            gnext += (uint32_t)(Kn * In * 4);
        }

        // ---- gates = bias + h*Whh^T + x*Wih^T  via V_WMMA_F32_16X16X4_F32
        v8f acc;
#pragma unroll
        for (int i = 0; i < 8; ++i) acc[i] = biasv;
#pragma unroll
        for (int c = 0; c < 5; ++c) {
            v2f a = *(const v2f*)&h_s[n * HS + 4 * c + kk0];
            acc = __builtin_amdgcn_wmma_f32_16x16x4_f32(
                      false, a, false, bfr[c], (short)0, acc, false, false);
        }
        {
            v2f ax = *(const v2f*)&x_s[t & 3][n][kk0];
            acc = __builtin_amdgcn_wmma_f32_16x16x4_f32(
                      false, ax, false, bx, (short)0, acc, false, false);
        }
#pragma unroll
        for (int r = 0; r < 8; ++r)
            g_s[(r + 8 * lh) * GS + wid * 16 + n] = acc[r];
        __syncthreads();   // gates complete

        // ---- elementwise cell update: c = s(f)*c + s(i)*tanh(g); h = s(o)*tanh(c)
        {
            float gi = g_s[rr0 * GS + hh0];
            float gf = g_s[rr0 * GS + 20 + hh0];
            float gg = g_s[rr0 * GS + 40 + hh0];
            float go = g_s[rr0 * GS + 60 + hh0];
            cst0 = fsig(gf) * cst0 + fsig(gi) * ftanh(gg);
            h_s[rr0 * HS + hh0] = fsig(go) * ftanh(cst0);
        }
        {
            float gi = g_s[rr1 * GS + hh1];
            float gf = g_s[rr1 * GS + 20 + hh1];
            float gg = g_s[rr1 * GS + 40 + hh1];
            float go = g_s[rr1 * GS + 60 + hh1];
            cst1 = fsig(gf) * cst1 + fsig(gi) * ftanh(gg);
            h_s[rr1 * HS + hh1] = fsig(go) * ftanh(cst1);
        }
    };

    // steady state: always 2 future steps in flight after the wait
    for (int t = 0; t < Tn - 3; ++t) {
        if (wid == 0) asm volatile("s_wait_asynccnt 0x6" ::: "memory");
        step(t, true);
    }
    // drain tail: t = Tn-3, Tn-2, Tn-1
    if (wid == 0) asm volatile("s_wait_asynccnt 0x6" ::: "memory");
    step(Tn - 3, false);
    if (wid == 0) asm volatile("s_wait_asynccnt 0x3" ::: "memory");
    step(Tn - 2, false);
    if (wid == 0) asm volatile("s_wait_asynccnt 0x0" ::: "memory");
    step(Tn - 1, false);

    __syncthreads();

    // ---- readout: y = tanh(h . conv_w[0:20] + w[1+kb]*conv_w[20] + conv_b)
    if (tid < 16) {
        int kb = k * Bsz + mt * 16 + tid;
        float a2 = conv_b[0] + wvec[1 + kb] * conv_w[20];
#pragma unroll
        for (int j = 0; j < Hn; ++j) a2 += h_s[tid * HS + j] * conv_w[j];
        logits[1 + kb] = ftanh(a2);
    }
    if (blockIdx.x == 0 && tid == 16) logits[0] = 1.0f;
}

__global__ __launch_bounds__(256) void softmax1025_kernel(
    const float* __restrict__ logits, float* __restrict__ out)
{
    __shared__ float red[256];
    const int tid = threadIdx.x;
    const int N = Kn * Bsz + 1;

    float m = -3.402823466e+38f;
    for (int i = tid; i < N; i += 256) m = fmaxf(m, logits[i]);
    red[tid] = m; __syncthreads();
    for (int s = 128; s > 0; s >>= 1) {
        if (tid < s) red[tid] = fmaxf(red[tid], red[tid + s]);
        __syncthreads();
    }
    const float mx = red[0]; __syncthreads();

    float sum = 0.f;
    for (int i = tid; i < N; i += 256) sum += __expf(logits[i] - mx);
    red[tid] = sum; __syncthreads();
    for (int s = 128; s > 0; s >>= 1) {
        if (tid < s) red[tid] += red[tid + s];
        __syncthreads();
    }
    const float inv = __builtin_amdgcn_rcpf(red[0]);
    for (int i = tid; i < N; i += 256) out[i] = __expf(logits[i] - mx) * inv;
}

extern "C" void kernel_launch(void* const* d_in, const int* in_sizes, int n_in,
                              void* d_out, int out_size, void* d_ws, size_t ws_size,
                              hipStream_t stream) {
    const float* inp    = (const float*)d_in[0];
    const float* wvec   = (const float*)d_in[1];
    const float* W_ih   = (const float*)d_in[2];
    const float* W_hh   = (const float*)d_in[3];
    const float* b_ih   = (const float*)d_in[4];
    const float* b_hh   = (const float*)d_in[5];
    const float* conv_w = (const float*)d_in[6];
    const float* conv_b = (const float*)d_in[7];

    float* logits = (float*)d_ws;                 // 1025 floats of scratch

    lstm16_wmma_kernel<<<Kn * 4, 160, 0, stream>>>(
        inp, wvec, W_ih, W_hh, b_ih, b_hh, conv_w, conv_b, logits);
    softmax1025_kernel<<<1, 256, 0, stream>>>(logits, (float*)d_out);
}